// PGAHead_42279658062165
// MI455X (gfx1250) — compile-verified
//
#include <hip/hip_runtime.h>
#include <math.h>

#define EPSV 1e-8f
#define BN_EPSV 1e-5f
#define BIGV 1e9f

typedef float v2f __attribute__((ext_vector_type(2)));
typedef float v8f __attribute__((ext_vector_type(8)));
typedef unsigned int u32x4 __attribute__((ext_vector_type(4)));
typedef int i32x4 __attribute__((ext_vector_type(4)));
typedef int i32x8 __attribute__((ext_vector_type(8)));

#if defined(__has_builtin)
#if __has_builtin(__builtin_amdgcn_tensor_load_to_lds) && \
    __has_builtin(__builtin_amdgcn_s_wait_tensorcnt)
#define USE_TDM 1
#endif
#endif
#ifndef USE_TDM
#define USE_TDM 0
#endif

#if USE_TDM
// DMA a 128-row x 32-float tile (row stride = row_stride_elems floats) from
// global memory into LDS at lds_tile, padding 2 dwords after every 32 dwords
// so the LDS row stride is 34 floats (matches LDT below, keeps v2f loads
// 8B-aligned). One TDM descriptor, tracked by TENSORcnt.
__device__ __forceinline__ void tdm_load_tile_128x32(float* lds_tile,
                                                     const float* gsrc,
                                                     int row_stride_elems) {
  unsigned long long ga = (unsigned long long)(uintptr_t)gsrc;
  u32x4 g0;
  g0.x = 1u;                                    // D# count = 1 (valid)
  g0.y = (unsigned)(uintptr_t)lds_tile;         // LDS byte address
  g0.z = (unsigned)ga;                          // global_addr[31:0]
  g0.w = (unsigned)((ga >> 32) & 0x01FFFFFFu)   // global_addr[56:32]
         | 0x80000000u;                         // type = 2 ("image") at [127:126]
  unsigned td0 = (unsigned)row_stride_elems;    // tensor_dim0 (>= tile width)
  const unsigned td1 = 128u;                    // tensor_dim1 (>= tile rows)
  i32x8 g1;
  g1[0] = (int)((2u << 16)                      // data_size = 4 bytes
                | (1u << 20)                    // pad_enable
                | (4u << 22)                    // pad_interval: every 32 dwords
                | (1u << 25));                  // pad_amount: 2 dwords
  g1[1] = (int)((td0 & 0xFFFFu) << 16);         // tensor_dim0[15:0]
  g1[2] = (int)((td0 >> 16) | (td1 << 16));     // tensor_dim0[31:16] | dim1 lo
  g1[3] = (int)(32u << 16);                     // dim1 hi | tile_dim0 = 32
  g1[4] = (int)128u;                            // tile_dim1 = 128 (tile_dim2 = 0)
  g1[5] = (int)(unsigned)row_stride_elems;      // tensor_dim0_stride[31:0]
  g1[6] = 0;
  g1[7] = 0;
  i32x4 gz = {0, 0, 0, 0};
#if defined(__clang_major__) && (__clang_major__ >= 23)
  i32x8 gz8 = {0, 0, 0, 0, 0, 0, 0, 0};
  __builtin_amdgcn_tensor_load_to_lds(g0, g1, gz, gz, gz8, 0);
#else
  __builtin_amdgcn_tensor_load_to_lds(g0, g1, gz, gz, 0);
#endif
}
#endif

// ---------------------------------------------------------------- reductions
__device__ __forceinline__ float block_reduce_sum(float v, float* red) {
  int tid = threadIdx.x;
  red[tid] = v; __syncthreads();
  for (int s = blockDim.x >> 1; s > 0; s >>= 1) {
    if (tid < s) red[tid] += red[tid + s];
    __syncthreads();
  }
  float r = red[0]; __syncthreads();
  return r;
}

// ---------------------------------------------------------------- WMMA GEMM
// C[M,N] = A[M,K] * B  (+ E if RESID, clip if CLIP)
// BT=false: B is row-major K x N.  BT=true: B is row-major N x K (i.e. A*B^T).
// Block tile 128x128x32, 256 threads = 8 waves, wave tile 32x64 (2x4 WMMA).
// A tiles (and B tiles when BT) are fetched by the Tensor Data Mover straight
// into LDS; NN-mode B tiles are transposed into LDS by the threads.
template<bool BT, bool CLIP, bool RESID>
__global__ __launch_bounds__(256)
void gemm_wmma_kernel(const float* __restrict__ A, const float* __restrict__ Bm,
                      const float* __restrict__ E, float* __restrict__ C,
                      int M, int N, int K, float clip_lo, float clip_hi) {
  const int BK = 32;
  const int LDT = BK + 2;                 // 34 floats: keeps v2f loads 8B aligned
  __shared__ __attribute__((aligned(16))) float As[128 * 34];
  __shared__ __attribute__((aligned(16))) float Bs[128 * 34];

  const int tid = threadIdx.x;
  const int lane = tid & 31;
  const int wave = tid >> 5;
  const int lane_lo = lane & 15;
  const int lane_hi = lane >> 4;
  const int wm = wave >> 1;               // 0..3
  const int wn = wave & 1;                // 0..1
  const int m0 = blockIdx.y * 128;
  const int n0 = blockIdx.x * 128;

  v8f acc[2][4];
#pragma unroll
  for (int mi = 0; mi < 2; ++mi)
#pragma unroll
    for (int ni = 0; ni < 4; ++ni)
#pragma unroll
      for (int r = 0; r < 8; ++r) acc[mi][ni][r] = 0.0f;

  const int ktiles = K / BK;
  for (int kt = 0; kt < ktiles; ++kt) {
    const int k0 = kt * BK;
#if USE_TDM
    if (wave == 0) {   // one DMA issue per tile (TDM ignores EXEC, per-wave)
      tdm_load_tile_128x32(As, A + (size_t)m0 * K + k0, K);
      if (BT) tdm_load_tile_128x32(Bs, Bm + (size_t)n0 * K + k0, K);
    }
    if (!BT) {
#pragma unroll
      for (int r = 0; r < 16; ++r) {
        int idx = tid + r * 256;
        int kk = idx >> 7, nn = idx & 127;
        Bs[nn * LDT + kk] = Bm[(size_t)(k0 + kk) * N + (n0 + nn)];
      }
      if (kt + 1 < ktiles)
        __builtin_prefetch(&Bm[(size_t)(k0 + BK + (tid >> 7)) * N + n0 + (tid & 127)], 0, 3);
    }
    if (wave == 0) __builtin_amdgcn_s_wait_tensorcnt(0);
#else
#pragma unroll
    for (int r = 0; r < 16; ++r) {
      int idx = tid + r * 256;
      int mm = idx >> 5, kk = idx & 31;
      As[mm * LDT + kk] = A[(size_t)(m0 + mm) * K + (k0 + kk)];
    }
#pragma unroll
    for (int r = 0; r < 16; ++r) {
      int idx = tid + r * 256;
      if (BT) {
        int nn = idx >> 5, kk = idx & 31;
        Bs[nn * LDT + kk] = Bm[(size_t)(n0 + nn) * K + (k0 + kk)];
      } else {
        int kk = idx >> 7, nn = idx & 127;
        Bs[nn * LDT + kk] = Bm[(size_t)(k0 + kk) * N + (n0 + nn)];
      }
    }
    if (kt + 1 < ktiles) {
      __builtin_prefetch(&A[(size_t)(m0 + (tid >> 5)) * K + (k0 + BK) + (tid & 31)], 0, 3);
      if (BT)
        __builtin_prefetch(&Bm[(size_t)(n0 + (tid >> 5)) * K + (k0 + BK) + (tid & 31)], 0, 3);
      else
        __builtin_prefetch(&Bm[(size_t)(k0 + BK + (tid >> 7)) * N + n0 + (tid & 127)], 0, 3);
    }
#endif
    __syncthreads();
#pragma unroll
    for (int ks = 0; ks < BK; ks += 4) {
      v2f afrag[2], bfrag[4];
#pragma unroll
      for (int mi = 0; mi < 2; ++mi) {
        int row = wm * 32 + mi * 16 + lane_lo;
        afrag[mi] = *(const v2f*)&As[row * LDT + ks + 2 * lane_hi];
      }
#pragma unroll
      for (int ni = 0; ni < 4; ++ni) {
        int col = wn * 64 + ni * 16 + lane_lo;
        bfrag[ni] = *(const v2f*)&Bs[col * LDT + ks + 2 * lane_hi];
      }
#pragma unroll
      for (int mi = 0; mi < 2; ++mi)
#pragma unroll
        for (int ni = 0; ni < 4; ++ni)
          acc[mi][ni] = __builtin_amdgcn_wmma_f32_16x16x4_f32(
              false, afrag[mi], false, bfrag[ni], (short)0, acc[mi][ni],
              false, false);
    }
    __syncthreads();
  }

#pragma unroll
  for (int mi = 0; mi < 2; ++mi)
#pragma unroll
    for (int ni = 0; ni < 4; ++ni)
#pragma unroll
      for (int r = 0; r < 8; ++r) {
        int row = m0 + wm * 32 + mi * 16 + lane_hi * 8 + r;
        int col = n0 + wn * 64 + ni * 16 + lane_lo;
        float v = acc[mi][ni][r];
        if (CLIP) v = fminf(fmaxf(v, clip_lo), clip_hi);
        size_t o = (size_t)row * N + col;
        if (RESID) v += E[o];
        C[o] = v;
      }
}

// ---------------------------------------------------------------- elementwise
__global__ __launch_bounds__(256)
void rownorm_kernel(const float* __restrict__ X, float* __restrict__ Y, int D) {
  __shared__ float red[256];
  int row = blockIdx.x, tid = threadIdx.x;
  const float* x = X + (size_t)row * D;
  float s = 0.f;
  for (int j = tid; j < D; j += 256) { float v = x[j]; s += v * v; }
  float tot = block_reduce_sum(s, red);
  float inv = 1.f / fmaxf(sqrtf(tot), EPSV);
  float* y = Y + (size_t)row * D;
  for (int j = tid; j < D; j += 256) y[j] = x[j] * inv;
}

__global__ void zero_kernel(float* p, int n) {
  int i = blockIdx.x * 256 + threadIdx.x;
  if (i < n) p[i] = 0.f;
}

// ---------------------------------------------------------------- top-k mask
__global__ __launch_bounds__(256)
void topk_kernel(const float* __restrict__ S, const int* __restrict__ labels,
                 unsigned* __restrict__ mrow, int Bn) {
  __shared__ float sv[2048];
  __shared__ int si[2048];
  __shared__ float redv[256];
  __shared__ int redi[256];
  __shared__ int selj[8];
  int row = blockIdx.x, tid = threadIdx.x;
  int li = labels[row];
  float tv[8]; int ti[8];
#pragma unroll
  for (int r = 0; r < 8; ++r) { tv[r] = -3.0e38f; ti[r] = -1; }
  const float* srow = S + (size_t)row * Bn;
  for (int j = tid; j < Bn; j += 256) {
    float m;
    if (labels[j] == li) m = (j == row) ? (srow[j] - BIGV) : srow[j];
    else m = -BIGV;
    if (m > tv[7]) {
      int p = 7;
      while (p > 0 && m > tv[p - 1]) { tv[p] = tv[p - 1]; ti[p] = ti[p - 1]; --p; }
      tv[p] = m; ti[p] = j;
    }
  }
#pragma unroll
  for (int r = 0; r < 8; ++r) { sv[tid * 8 + r] = tv[r]; si[tid * 8 + r] = ti[r]; }
  __syncthreads();
  for (int p = 0; p < 8; ++p) {
    float bm = -3.0e38f; int bs = 0;
#pragma unroll
    for (int r = 0; r < 8; ++r) {
      float v = sv[tid * 8 + r];
      if (v > bm) { bm = v; bs = tid * 8 + r; }
    }
    redv[tid] = bm; redi[tid] = bs; __syncthreads();
    for (int s = 128; s > 0; s >>= 1) {
      if (tid < s && redv[tid + s] > redv[tid]) {
        redv[tid] = redv[tid + s]; redi[tid] = redi[tid + s];
      }
      __syncthreads();
    }
    if (tid == 0) { int slot = redi[0]; selj[p] = si[slot]; sv[slot] = -3.0e38f; }
    __syncthreads();
  }
  unsigned* mr = mrow + (size_t)row * (Bn / 32);
  for (int w = tid; w < Bn / 32; w += 256) mr[w] = 0u;
  __syncthreads();
  if (tid == 0) {
    for (int p = 0; p < 8; ++p) {
      int j = selj[p];
      if (j >= 0 && j != row && labels[j] == li) mr[j >> 5] |= (1u << (j & 31));
    }
  }
}

__global__ __launch_bounds__(128)
void symm_mask_kernel(const unsigned* __restrict__ mrow, const int* __restrict__ labels,
                      unsigned* __restrict__ Mb, int Bn) {
  int i = blockIdx.x, w = threadIdx.x;
  int W = Bn / 32;
  unsigned mi = mrow[(size_t)i * W + w];
  int li = labels[i];
  unsigned out = 0u;
  for (int b = 0; b < 32; ++b) {
    int j = w * 32 + b;
    unsigned mij = (mi >> b) & 1u;
    unsigned mji = (mrow[(size_t)j * W + (i >> 5)] >> (i & 31)) & 1u;
    if ((mij | mji) && (labels[j] == li) && (j != i)) out |= (1u << b);
  }
  Mb[(size_t)i * W + w] = out;
}

__global__ __launch_bounds__(256)
void degree_kernel(const float* __restrict__ S, const unsigned* __restrict__ Mb,
                   float* __restrict__ dinv, int Bn) {
  __shared__ float red[256];
  int i = blockIdx.x, tid = threadIdx.x;
  const float* srow = S + (size_t)i * Bn;
  const unsigned* mr = Mb + (size_t)i * (Bn / 32);
  float s = 0.f;
  for (int j = tid; j < Bn; j += 256)
    if ((mr[j >> 5] >> (j & 31)) & 1u) s += fmaxf(srow[j], 0.f);
  float tot = block_reduce_sum(s, red);
  if (tid == 0) {
    float d = fmaxf(tot + 1e-6f, EPSV);
    dinv[i] = 1.0f / sqrtf(d);
  }
}

__global__ __launch_bounds__(256)
void formK_kernel(float* __restrict__ S, const unsigned* __restrict__ Mb,
                  const float* __restrict__ dinv, int Bn) {
  int i = blockIdx.y;
  int j = blockIdx.x * 256 + threadIdx.x;
  size_t o = (size_t)i * Bn + j;
  float a = ((Mb[(size_t)i * (Bn / 32) + (j >> 5)] >> (j & 31)) & 1u)
                ? fmaxf(S[o], 0.f) : 0.f;
  if (i == j) a += 1e-6f;
  S[o] = dinv[i] * a * dinv[j];
}

// ---------------------------------------------------------------- batchnorm
__global__ __launch_bounds__(256)
void bnsum_kernel(const float* __restrict__ Y, float* __restrict__ s1,
                  float* __restrict__ s2, int D) {
  int r0 = blockIdx.x * 32;
  for (int c = threadIdx.x; c < D; c += 256) {
    float a = 0.f, b = 0.f;
    for (int r = 0; r < 32; ++r) {
      float v = Y[(size_t)(r0 + r) * D + c];
      a += v; b += v * v;
    }
    atomicAdd(&s1[c], a);
    atomicAdd(&s2[c], b);
  }
}

__global__ void bnfin_kernel(const float* __restrict__ s1, const float* __restrict__ s2,
                             float* __restrict__ mu, float* __restrict__ rsig,
                             int Bn, int D) {
  int c = blockIdx.x * 256 + threadIdx.x;
  if (c < D) {
    float m = s1[c] / (float)Bn;
    float v = s2[c] / (float)Bn - m * m;
    mu[c] = m;
    rsig[c] = rsqrtf(v + BN_EPSV);
  }
}

__global__ __launch_bounds__(256)
void bnapply_kernel(float* __restrict__ Y, const float* __restrict__ mu,
                    const float* __restrict__ rsig, const float* __restrict__ gamma,
                    const float* __restrict__ beta, int D) {
  int r = blockIdx.y;
  int c = blockIdx.x * 256 + threadIdx.x;
  size_t o = (size_t)r * D + c;
  float z = gamma[c] * (Y[o] - mu[c]) * rsig[c] + beta[c];
  Y[o] = fmaxf(z, 0.f);
}

// ---------------------------------------------------------------- losses
__global__ __launch_bounds__(256)
void lossK_kernel(const float* __restrict__ Kp, const float* __restrict__ Kc,
                  const unsigned* __restrict__ Mp, const unsigned* __restrict__ Mc,
                  float* __restrict__ acc, int Bn) {
  __shared__ float red[256];
  int W = Bn / 32;
  int w = blockIdx.x * 256 + threadIdx.x;
  float num = 0.f, den = 0.f;
  if (w < Bn * W) {
    unsigned bits = Mp[w] | Mc[w];
    den = (float)__popc(bits);
    int i = w / W;
    int jc = (w % W) * 32;
    while (bits) {
      int b = __ffs(bits) - 1;
      bits &= bits - 1;
      size_t o = (size_t)i * Bn + (jc + b);
      float d = Kp[o] - Kc[o];
      num += d * d;
    }
  }
  float tn = block_reduce_sum(num, red);
  float td = block_reduce_sum(den, red);
  if (threadIdx.x == 0) { atomicAdd(&acc[0], tn); atomicAdd(&acc[1], td); }
}

__global__ __launch_bounds__(256)
void lossZ_kernel(const float* __restrict__ Zp, const float* __restrict__ Zc,
                  float* __restrict__ acc, int n) {
  __shared__ float red[256];
  int i = blockIdx.x * 256 + threadIdx.x;
  float s = 0.f;
  if (i < n) { float d = Zp[i] - Zc[i]; s = d * d; }
  float t = block_reduce_sum(s, red);
  if (threadIdx.x == 0) atomicAdd(acc, t);
}

__global__ void finalize_kernel(const float* __restrict__ acc, float* __restrict__ out,
                                int pairs, float zcount) {
  if (threadIdx.x == 0 && blockIdx.x == 0) {
    float lk = 0.f, lz = 0.f;
    for (int p = 0; p < pairs; ++p) {
      lk += acc[2 * p] / fmaxf(acc[2 * p + 1], EPSV);
      lz += acc[6 + p] / zcount;
    }
    out[0] = lk;
    out[1] = lz;
    out[2] = 64.0f * lk + 16.0f * lz;
  }
}

// ---------------------------------------------------------------- launch
extern "C" void kernel_launch(void* const* d_in, const int* in_sizes, int n_in,
                              void* d_out, int out_size, void* d_ws, size_t ws_size,
                              hipStream_t stream) {
  (void)in_sizes; (void)n_in; (void)out_size; (void)ws_size;
  const int L = 4, Bn = 4096, D = 512, P = 768, W = Bn / 32;
  const float* feats = (const float*)d_in[0];
  const int* labels = (const int*)d_in[1];
  const float* fc1 = (const float*)d_in[2];
  const float* fc2 = (const float*)d_in[3];
  const float* bng = (const float*)d_in[4];
  const float* bnb = (const float*)d_in[5];
  const float* projw = (const float*)d_in[6];

  char* wsp = (char*)d_ws;
  auto alloc = [&](size_t bytes) {
    char* p = wsp;
    wsp += (bytes + 255) & ~(size_t)255;
    return p;
  };
  float* Kbuf[2] = { (float*)alloc((size_t)Bn * Bn * 4), (float*)alloc((size_t)Bn * Bn * 4) };
  unsigned* Mbuf[2] = { (unsigned*)alloc((size_t)Bn * W * 4), (unsigned*)alloc((size_t)Bn * W * 4) };
  unsigned* mrow = (unsigned*)alloc((size_t)Bn * W * 4);
  float* Xn = (float*)alloc((size_t)Bn * D * 4);
  float* H1 = (float*)alloc((size_t)Bn * D * 4);
  float* Y1 = (float*)alloc((size_t)Bn * D * 4);
  float* H2 = (float*)alloc((size_t)Bn * D * 4);
  float* Zb = (float*)alloc((size_t)Bn * D * 4);
  float* ZPb[2] = { (float*)alloc((size_t)Bn * P * 4), (float*)alloc((size_t)Bn * P * 4) };
  float* ZPpre = (float*)alloc((size_t)Bn * P * 4);
  float* dinv = (float*)alloc((size_t)Bn * 4);
  float* bns1 = (float*)alloc((size_t)2 * D * 4);
  float* bns2 = bns1 + D;
  float* bmu = (float*)alloc((size_t)D * 4);
  float* brs = (float*)alloc((size_t)D * 4);
  float* acc = (float*)alloc(64 * 4);

  dim3 thr(256);
  const float clo = (float)(-1.0 + 1e-8);
  const float chi = (float)(1.0 - 1e-8);

  zero_kernel<<<1, 256, 0, stream>>>(acc, 9);

  for (int i = 0; i < L; ++i) {
    const float* X = feats + (size_t)i * Bn * D;
    float* Kc = Kbuf[i & 1];
    unsigned* Mc = Mbuf[i & 1];
    float* ZPc = ZPb[i & 1];

    // graph: cosine sim -> knn mask -> normalized adjacency K (in place)
    rownorm_kernel<<<Bn, thr, 0, stream>>>(X, Xn, D);
    gemm_wmma_kernel<true, true, false><<<dim3(Bn / 128, Bn / 128), thr, 0, stream>>>(
        Xn, Xn, nullptr, Kc, Bn, Bn, D, clo, chi);
    topk_kernel<<<Bn, thr, 0, stream>>>(Kc, labels, mrow, Bn);
    symm_mask_kernel<<<Bn, dim3(W), 0, stream>>>(mrow, labels, Mc, Bn);
    degree_kernel<<<Bn, thr, 0, stream>>>(Kc, Mc, dinv, Bn);
    formK_kernel<<<dim3(Bn / 256, Bn), thr, 0, stream>>>(Kc, Mc, dinv, Bn);

    // GAM: Z = K @ relu(BN(K @ (X w1))) w2 + X
    gemm_wmma_kernel<false, false, false><<<dim3(D / 128, Bn / 128), thr, 0, stream>>>(
        X, fc1 + (size_t)i * D * D, nullptr, H1, Bn, D, D, 0.f, 0.f);
    gemm_wmma_kernel<false, false, false><<<dim3(D / 128, Bn / 128), thr, 0, stream>>>(
        Kc, H1, nullptr, Y1, Bn, D, Bn, 0.f, 0.f);
    zero_kernel<<<(2 * D + 255) / 256, thr, 0, stream>>>(bns1, 2 * D);
    bnsum_kernel<<<Bn / 32, thr, 0, stream>>>(Y1, bns1, bns2, D);
    bnfin_kernel<<<(D + 255) / 256, thr, 0, stream>>>(bns1, bns2, bmu, brs, Bn, D);
    bnapply_kernel<<<dim3(D / 256, Bn), thr, 0, stream>>>(
        Y1, bmu, brs, bng + (size_t)i * D, bnb + (size_t)i * D, D);
    gemm_wmma_kernel<false, false, false><<<dim3(D / 128, Bn / 128), thr, 0, stream>>>(
        Y1, fc2 + (size_t)i * D * D, nullptr, H2, Bn, D, D, 0.f, 0.f);
    gemm_wmma_kernel<false, false, true><<<dim3(D / 128, Bn / 128), thr, 0, stream>>>(
        Kc, H2, X, Zb, Bn, D, Bn, 0.f, 0.f);

    // projection + row-normalize
    gemm_wmma_kernel<false, false, false><<<dim3(P / 128, Bn / 128), thr, 0, stream>>>(
        Zb, projw, nullptr, ZPpre, Bn, P, D, 0.f, 0.f);
    rownorm_kernel<<<Bn, thr, 0, stream>>>(ZPpre, ZPc, P);

    if (i > 0) {
      float* Kp = Kbuf[(i - 1) & 1];
      unsigned* Mp = Mbuf[(i - 1) & 1];
      float* ZPp = ZPb[(i - 1) & 1];
      lossK_kernel<<<(Bn * W) / 256, thr, 0, stream>>>(Kp, Kc, Mp, Mc, acc + 2 * (i - 1), Bn);
      lossZ_kernel<<<(Bn * P) / 256, thr, 0, stream>>>(ZPp, ZPc, acc + 6 + (i - 1), Bn * P);
    }
  }
  finalize_kernel<<<1, 32, 0, stream>>>(acc, (float*)d_out, L - 1, (float)((size_t)Bn * P));
}